// PromptCompressorLightweight_53025666237215
// MI455X (gfx1250) — compile-verified
//
#include <hip/hip_runtime.h>
#include <hip/hip_bf16.h>

// Shapes (fixed by the reference)
#define BB 1
#define HH 8
#define HQ 32
#define TT 2048
#define DD 128
#define MM 4096
#define FF 5
#define HID 10
#define LL 512

typedef __attribute__((ext_vector_type(2))) float v2f;
typedef __attribute__((ext_vector_type(8))) float v8f;

__device__ __forceinline__ float wave_reduce_sum(float v) {
    // wave32 butterfly reduction
    #pragma unroll
    for (int off = 16; off > 0; off >>= 1) v += __shfl_xor(v, off, 32);
    return v;
}

// ---------------------------------------------------------------------------
// emb_norm[t] = ||x[t,:]|| over M=4096; write into feats[h][t][4] for all h.
// One wave per token. grid.x = T/8, block = 256 (8 waves).
// ---------------------------------------------------------------------------
__global__ __launch_bounds__(256) void emb_norm_kernel(const float* __restrict__ x,
                                                       float* __restrict__ feats) {
    int wave = threadIdx.x >> 5, lane = threadIdx.x & 31;
    int t = blockIdx.x * 8 + wave;
    const float4* xr = (const float4*)(x + (size_t)t * MM);
    float s = 0.f;
    #pragma unroll 4
    for (int j = 0; j < MM / 4 / 32; ++j) {
        float4 v = xr[lane + 32 * j];
        s += v.x * v.x + v.y * v.y + v.z * v.z + v.w * v.w;
    }
    s = wave_reduce_sum(s);
    if (lane == 0) {
        float n = sqrtf(s);
        #pragma unroll
        for (int h = 0; h < HH; ++h)
            feats[((size_t)h * TT + t) * FF + 4] = n;
    }
}

// ---------------------------------------------------------------------------
// key/value/query norms. One wave per (h,t). grid.x = H*T/8, block = 256.
// feats[h][t][1]=||k||, [2]=||v||, [3]=||mean_g q||
// ---------------------------------------------------------------------------
__global__ __launch_bounds__(256) void norms_kernel(const float* __restrict__ k_val,
                                                    const float* __restrict__ v_val,
                                                    const float* __restrict__ query,
                                                    float* __restrict__ feats) {
    int wave = threadIdx.x >> 5, lane = threadIdx.x & 31;
    int task = blockIdx.x * 8 + wave;
    int h = task >> 11;        // / T
    int t = task & (TT - 1);

    // key norm: 128 floats = 32 lanes * float4
    const float4* kb = (const float4*)(k_val + ((size_t)h * TT + t) * DD);
    float4 kv = kb[lane];
    float ks = wave_reduce_sum(kv.x * kv.x + kv.y * kv.y + kv.z * kv.z + kv.w * kv.w);

    const float4* vb = (const float4*)(v_val + ((size_t)h * TT + t) * DD);
    float4 vv = vb[lane];
    float vs = wave_reduce_sum(vv.x * vv.x + vv.y * vv.y + vv.z * vv.z + vv.w * vv.w);

    // q_avg over 4 query heads, then norm
    float4 qa = make_float4(0.f, 0.f, 0.f, 0.f);
    #pragma unroll
    for (int g = 0; g < 4; ++g) {
        const float4* qb = (const float4*)(query + ((size_t)(h * 4 + g) * TT + t) * DD);
        float4 q = qb[lane];
        qa.x += q.x; qa.y += q.y; qa.z += q.z; qa.w += q.w;
    }
    qa.x *= 0.25f; qa.y *= 0.25f; qa.z *= 0.25f; qa.w *= 0.25f;
    float qs = wave_reduce_sum(qa.x * qa.x + qa.y * qa.y + qa.z * qa.z + qa.w * qa.w);

    if (lane == 0) {
        float* f = feats + ((size_t)h * TT + t) * FF;
        f[1] = sqrtf(ks);
        f[2] = sqrtf(vs);
        f[3] = sqrtf(qs);
    }
}

// ---------------------------------------------------------------------------
// attn_score[h][j] = (sum_i attn[h,i,j]*(i+1)) / (T - j)  via WMMA f32 16x16x4.
// Broadcast-A trick: A[m,k] = w[i+k] for all m, B[k,n] = attn[i+k, j0+n].
// Every D row is the identical weighted column sum; read it from row M=0
// (lanes 0-15, VGPR0). Row weights carried as a float accumulator (+8.0f/step,
// exact for i<=2048) so the hot loop is just NT loads + WMMA + 2 fadds.
// attn is streamed once -> non-temporal loads keep L2 free for k/v reuse.
// grid = (T/16/8, H), block = 256 (8 waves).
// ---------------------------------------------------------------------------
__global__ __launch_bounds__(256) void attn_score_kernel(const float* __restrict__ attn,
                                                         float* __restrict__ feats) {
    int wave = threadIdx.x >> 5, lane = threadIdx.x & 31;
    int h = blockIdx.y;
    int j0 = (blockIdx.x * 8 + wave) * 16;
    int half = lane >> 4;       // 0: B rows K=0,1  |  1: B rows K=2,3
    int col = lane & 15;        // column within tile -> lane

    const float* ah = attn + (size_t)h * TT * TT + (j0 + col);

    float base = (float)(half * 2 + 1);   // weight of this lane's first B row
    v8f c0 = {};
    v8f c1 = {};
    for (int i = 0; i < TT; i += 8) {
        size_t r0 = (size_t)(i + half * 2);
        v2f a0, b0;
        b0.x = __builtin_nontemporal_load(ah + r0 * TT);
        b0.y = __builtin_nontemporal_load(ah + (r0 + 1) * TT);
        a0.x = base;
        a0.y = base + 1.0f;
        c0 = __builtin_amdgcn_wmma_f32_16x16x4_f32(false, a0, false, b0,
                                                   (short)0, c0, false, false);
        v2f a1, b1;
        b1.x = __builtin_nontemporal_load(ah + (r0 + 4) * TT);
        b1.y = __builtin_nontemporal_load(ah + (r0 + 5) * TT);
        a1.x = base + 4.0f;
        a1.y = base + 5.0f;
        c1 = __builtin_amdgcn_wmma_f32_16x16x4_f32(false, a1, false, b1,
                                                   (short)0, c1, false, false);
        base += 8.0f;
    }
    if (half == 0) {
        int j = j0 + col;
        float s = c0[0] + c1[0];            // row M=0 of each accumulator
        feats[((size_t)h * TT + j) * FF + 0] = s / (float)(TT - j);
    }
}

// ---------------------------------------------------------------------------
// Tiny MLP: hidden = relu(feats @ W1[h] + b1[h]); priority = hidden @ W2[h] + b2[h]
// One thread per (h,t).
// ---------------------------------------------------------------------------
__global__ __launch_bounds__(256) void mlp_kernel(const float* __restrict__ feats,
                                                  const float* __restrict__ W1,
                                                  const float* __restrict__ b1,
                                                  const float* __restrict__ W2,
                                                  const float* __restrict__ b2,
                                                  float* __restrict__ priority) {
    int tid = blockIdx.x * blockDim.x + threadIdx.x;   // 0 .. H*T-1
    int h = tid >> 11;
    const float* f = feats + (size_t)tid * FF;
    float fr[FF];
    #pragma unroll
    for (int i = 0; i < FF; ++i) fr[i] = f[i];

    const float* w1 = W1 + (size_t)h * FF * HID;
    const float* bb1 = b1 + (size_t)h * HID;
    const float* w2 = W2 + (size_t)h * HID;
    float p = b2[h];
    #pragma unroll
    for (int k = 0; k < HID; ++k) {
        float acc = bb1[k];
        #pragma unroll
        for (int i = 0; i < FF; ++i) acc += fr[i] * w1[i * HID + k];
        acc = fmaxf(acc, 0.f);
        p += acc * w2[k];
    }
    priority[tid] = p;
}

// ---------------------------------------------------------------------------
// Per-head top-512 of 2048, tie-break = lowest index, output indices sorted
// ascending. Bitonic sort of packed u64 keys in LDS. One block (1024 thr)/head.
// ---------------------------------------------------------------------------
__global__ __launch_bounds__(1024) void topk_kernel(const float* __restrict__ priority,
                                                    int* __restrict__ keep) {
    __shared__ unsigned long long keys[TT];
    __shared__ int sidx[LL];
    int h = blockIdx.x, tid = threadIdx.x;

    for (int i = tid; i < TT; i += 1024) {
        unsigned u = __float_as_uint(priority[(size_t)h * TT + i]);
        unsigned s = (u & 0x80000000u) ? ~u : (u | 0x80000000u); // ascending-sortable
        keys[i] = ((unsigned long long)(~s) << 32) | (unsigned)i;  // desc prio, asc idx
    }
    __syncthreads();

    // bitonic sort ascending over 2048 keys
    for (unsigned k = 2; k <= TT; k <<= 1) {
        for (unsigned j = k >> 1; j > 0; j >>= 1) {
            for (unsigned i = tid; i < TT; i += 1024) {
                unsigned ixj = i ^ j;
                if (ixj > i) {
                    unsigned long long a = keys[i], b = keys[ixj];
                    bool up = ((i & k) == 0);
                    if ((a > b) == up) { keys[i] = b; keys[ixj] = a; }
                }
            }
            __syncthreads();
        }
    }

    // first 512 = top-512; extract indices
    for (int i = tid; i < LL; i += 1024) sidx[i] = (int)(keys[i] & 0xffffffffu);
    __syncthreads();

    // bitonic sort ascending over 512 ints
    for (unsigned k = 2; k <= LL; k <<= 1) {
        for (unsigned j = k >> 1; j > 0; j >>= 1) {
            for (unsigned i = tid; i < LL; i += 1024) {
                unsigned ixj = i ^ j;
                if (ixj > i) {
                    int a = sidx[i], b = sidx[ixj];
                    bool up = ((i & k) == 0);
                    if ((a > b) == up) { sidx[i] = b; sidx[ixj] = a; }
                }
            }
            __syncthreads();
        }
    }

    for (int i = tid; i < LL; i += 1024) keep[(size_t)h * LL + i] = sidx[i];
}

// ---------------------------------------------------------------------------
// Gather k/v rows with float4 (128-bit) transfers: one wave per (h,l) row,
// each lane moves one float4 of k and one of v. grid.x = H*L/8, block = 256.
// ---------------------------------------------------------------------------
__global__ __launch_bounds__(256) void gather_kernel(const float* __restrict__ k_val,
                                                     const float* __restrict__ v_val,
                                                     const int* __restrict__ keep,
                                                     float* __restrict__ k_out,
                                                     float* __restrict__ v_out) {
    int wave = threadIdx.x >> 5, lane = threadIdx.x & 31;
    int row = blockIdx.x * 8 + wave;      // 0 .. H*L-1
    int h = row >> 9;                     // / L
    int idx = keep[row];

    const float4* ks = (const float4*)(k_val + ((size_t)h * TT + idx) * DD);
    const float4* vs = (const float4*)(v_val + ((size_t)h * TT + idx) * DD);
    float4*       kd = (float4*)(k_out + (size_t)row * DD);
    float4*       vd = (float4*)(v_out + (size_t)row * DD);
    kd[lane] = ks[lane];
    vd[lane] = vs[lane];
}

extern "C" void kernel_launch(void* const* d_in, const int* in_sizes, int n_in,
                              void* d_out, int out_size, void* d_ws, size_t ws_size,
                              hipStream_t stream) {
    // inputs (setup_inputs order)
    const float* k_val = (const float*)d_in[1];
    const float* v_val = (const float*)d_in[2];
    const float* query = (const float*)d_in[3];
    const float* x     = (const float*)d_in[4];
    const float* attn  = (const float*)d_in[5];
    const float* W1    = (const float*)d_in[6];
    const float* b1    = (const float*)d_in[7];
    const float* W2    = (const float*)d_in[8];
    const float* b2    = (const float*)d_in[9];

    // workspace: feats[H][T][5] then priority[H][T]
    float* feats    = (float*)d_ws;
    float* priority = feats + (size_t)HH * TT * FF;

    // outputs concatenated: keep_idxs (H*L int32), k_out, v_out (each H*L*D f32)
    int*   keep  = (int*)d_out;
    float* k_out = (float*)d_out + (size_t)HH * LL;
    float* v_out = k_out + (size_t)HH * LL * DD;

    emb_norm_kernel<<<dim3(TT / 8), dim3(256), 0, stream>>>(x, feats);
    norms_kernel<<<dim3(HH * TT / 8), dim3(256), 0, stream>>>(k_val, v_val, query, feats);
    attn_score_kernel<<<dim3(TT / 16 / 8, HH), dim3(256), 0, stream>>>(attn, feats);
    mlp_kernel<<<dim3(HH * TT / 256), dim3(256), 0, stream>>>(feats, W1, b1, W2, b2, priority);
    topk_kernel<<<dim3(HH), dim3(1024), 0, stream>>>(priority, keep);
    gather_kernel<<<dim3(HH * LL / 8), dim3(256), 0, stream>>>(k_val, v_val, keep, k_out, v_out);
}